// RankingLoss_43516608643236
// MI455X (gfx1250) — compile-verified
//
#include <hip/hip_runtime.h>

// D = A(16x4 f32) * B(4x16 f32) + C(16x16 f32), wave32 WMMA.
typedef float v2f __attribute__((ext_vector_type(2)));
typedef float v8f __attribute__((ext_vector_type(8)));

#define NEG_PENALTY 0.03f
#define C_TOTAL     1000
#define C_LOADED    256
#define ROWS_PER_BLOCK 8
#define BLOCK_THREADS  (ROWS_PER_BLOCK * 32)

// One wave32 per row b. Stage gathered ranks/labels in LDS, then for each
// 16-wide p-tile accumulate sum_n relu(0.03 + u[n] - r_p) via
// V_WMMA_F32_16X16X4_F32 with an all-ones B operand (exact f32 reduction on
// the matrix pipe). u[n] = -1e30 for positive n folds the neg-mask into relu.
__global__ __launch_bounds__(BLOCK_THREADS)
void rank_loss_rows(const float* __restrict__ ranks,
                    const int*   __restrict__ labels,
                    const long long* __restrict__ ids,
                    float* __restrict__ row_total,
                    float* __restrict__ row_valid,
                    int B)
{
    __shared__ float s_rs [ROWS_PER_BLOCK][C_LOADED]; // gathered ranks
    __shared__ float s_u  [ROWS_PER_BLOCK][C_LOADED]; // neg ? r : -1e30
    __shared__ float s_pos[ROWS_PER_BLOCK][C_LOADED]; // label as float

    const int lane = threadIdx.x & 31;
    const int wid  = threadIdx.x >> 5;
    const int b    = blockIdx.x * ROWS_PER_BLOCK + wid;
    const bool valid = (b < B);
    const int brow = valid ? b : 0;
    const size_t rowbase = (size_t)brow * C_TOTAL;

    int anypos = 0;
    #pragma unroll
    for (int i = 0; i < C_LOADED / 32; ++i) {
        const int j = lane + 32 * i;
        const long long id = ids[j];
        const float r = ranks[rowbase + (size_t)id];
        const int   l = labels[rowbase + (size_t)id];
        s_rs [wid][j] = r;
        s_pos[wid][j] = (l != 0) ? 1.0f : 0.0f;
        s_u  [wid][j] = (l != 0) ? -1.0e30f : r;   // masked-out n never passes relu
        anypos |= l;
    }
    __syncthreads();

    // A-fragment layout (16x4 f32): lane L holds M = L%16; K-pair = 2*(L/16).
    const int m     = lane & 15;
    const int kbase = (lane >> 4) << 1;   // 0 or 2
    const int moff  = (lane >> 4) << 3;   // C rows: M = v (+8 for lanes 16..31)
    const v2f bones = {1.0f, 1.0f};

    float acc = 0.0f;
    for (int pt = 0; pt < C_LOADED / 16; ++pt) {
        const float c0 = NEG_PENALTY - s_rs[wid][pt * 16 + m];
        v8f c = {0.f, 0.f, 0.f, 0.f, 0.f, 0.f, 0.f, 0.f};
        #pragma unroll 8
        for (int nc = 0; nc < C_LOADED / 4; ++nc) {
            const float u0 = s_u[wid][nc * 4 + kbase + 0];
            const float u1 = s_u[wid][nc * 4 + kbase + 1];
            v2f a;
            a.x = fmaxf(c0 + u0, 0.0f);
            a.y = fmaxf(c0 + u1, 0.0f);
            c = __builtin_amdgcn_wmma_f32_16x16x4_f32(
                    /*neg_a=*/false, a, /*neg_b=*/false, bones,
                    /*c_mod=*/(short)0, c, /*reuse_a=*/false, /*reuse_b=*/false);
        }
        // Apply pos-mask per p row; every column of C holds the same row sum.
        #pragma unroll
        for (int v = 0; v < 8; ++v) {
            const int prow = pt * 16 + v + moff;
            acc += c[v] * s_pos[wid][prow];
        }
    }

    // Wave reduction (fixed order -> deterministic), columns replicated 16x.
    for (int off = 16; off > 0; off >>= 1) {
        acc    += __shfl_xor(acc, off, 32);
        anypos |= __shfl_xor(anypos, off, 32);
    }
    if (lane == 0 && valid) {
        row_total[b] = acc * (1.0f / 16.0f);
        row_valid[b] = anypos ? 1.0f : 0.0f;
    }
}

// Deterministic fixed-order final reduction + divide.
__global__ __launch_bounds__(256)
void final_reduce(const float* __restrict__ row_total,
                  const float* __restrict__ row_valid,
                  float* __restrict__ out, int B)
{
    __shared__ float st[256];
    __shared__ float sv[256];
    float t = 0.0f, v = 0.0f;
    for (int i = threadIdx.x; i < B; i += 256) {
        t += row_total[i];
        v += row_valid[i];
    }
    st[threadIdx.x] = t;
    sv[threadIdx.x] = v;
    __syncthreads();
    for (int s = 128; s > 0; s >>= 1) {
        if (threadIdx.x < s) {
            st[threadIdx.x] += st[threadIdx.x + s];
            sv[threadIdx.x] += sv[threadIdx.x + s];
        }
        __syncthreads();
    }
    if (threadIdx.x == 0) out[0] = st[0] / sv[0];
}

extern "C" void kernel_launch(void* const* d_in, const int* in_sizes, int n_in,
                              void* d_out, int out_size, void* d_ws, size_t ws_size,
                              hipStream_t stream)
{
    const float*      ranks  = (const float*)d_in[0];
    const int*        labels = (const int*)d_in[1];
    const long long*  ids    = (const long long*)d_in[2];

    const int B = in_sizes[0] / C_TOTAL;   // 2048

    float* row_total = (float*)d_ws;       // B floats
    float* row_valid = row_total + B;      // B floats

    const int grid = (B + ROWS_PER_BLOCK - 1) / ROWS_PER_BLOCK;
    rank_loss_rows<<<grid, BLOCK_THREADS, 0, stream>>>(
        ranks, labels, ids, row_total, row_valid, B);
    final_reduce<<<1, 256, 0, stream>>>(row_total, row_valid, (float*)d_out, B);
}